// Pointer_Generator_73023033966843
// MI455X (gfx1250) — compile-verified
//
#include <hip/hip_runtime.h>
#include <hip/hip_bf16.h>

// Problem constants (match reference setup_inputs exactly)
#define B_  4
#define T_  256
#define S_  512
#define H_  12
#define D_  768
#define V_  32000
#define BT_ (B_ * T_)

typedef float v2f __attribute__((ext_vector_type(2)));
typedef float v4f __attribute__((ext_vector_type(4)));
typedef float v8f __attribute__((ext_vector_type(8)));

// ---------------------------------------------------------------------------
// Kernel 1: p_gen via V_WMMA_F32_16X16X4_F32.
// X = dec_output viewed as (BT_, D_). One wave (32 threads) per 16 rows.
// A-tile: 16x4 chunk of X (lanes 0-15: K=0,1; lanes 16-31: K=2,3).
// B-tile: 4x16 with W_pgen chunk broadcast across all 16 columns, so every
// column of the 16x16 accumulator holds the same dot products.
// f32 C/D layout: VGPR v, lanes 0-15 -> M=v, lanes 16-31 -> M=8+v.
// ---------------------------------------------------------------------------
__global__ __launch_bounds__(32) void pgen_wmma_kernel(
    const float* __restrict__ X, const float* __restrict__ W,
    const float* __restrict__ bias, float* __restrict__ pg) {
  const int lane = threadIdx.x;        // 0..31
  const int half = lane >> 4;          // 0 or 1
  const int m    = lane & 15;          // row within tile
  const int r0   = blockIdx.x * 16;    // first row of this tile

  const float* xrow = X + (size_t)(r0 + m) * D_ + 2 * half;

  v8f acc = {};
  for (int k = 0; k < D_; k += 4) {
    v2f a, b;
    a.x = xrow[k];
    a.y = xrow[k + 1];
    b.x = W[k + 2 * half];
    b.y = W[k + 2 * half + 1];
    acc = __builtin_amdgcn_wmma_f32_16x16x4_f32(
        /*neg_a=*/false, a, /*neg_b=*/false, b,
        /*c_mod=*/(short)0, acc, /*reuse_a=*/false, /*reuse_b=*/false);
  }

  const float bb = bias[0];
  if (m == 0) {
    // lane 0 -> rows r0+0..7, lane 16 -> rows r0+8..15
    #pragma unroll
    for (int v = 0; v < 8; ++v) {
      float d = acc[v] + bb;
      pg[r0 + half * 8 + v] = 1.0f / (1.0f + __expf(-d));
    }
  }
}

// ---------------------------------------------------------------------------
// Block-wide reductions over 256 threads using an LDS scratch array.
// ---------------------------------------------------------------------------
__device__ inline float blockSum256(float v, float* red, int tid) {
  red[tid] = v;
  __syncthreads();
  #pragma unroll
  for (int off = 128; off > 0; off >>= 1) {
    if (tid < off) red[tid] += red[tid + off];
    __syncthreads();
  }
  float r = red[0];
  __syncthreads();
  return r;
}

__device__ inline float blockMax256(float v, float* red, int tid) {
  red[tid] = v;
  __syncthreads();
  #pragma unroll
  for (int off = 128; off > 0; off >>= 1) {
    if (tid < off) red[tid] = fmaxf(red[tid], red[tid + off]);
    __syncthreads();
  }
  float r = red[0];
  __syncthreads();
  return r;
}

// ---------------------------------------------------------------------------
// Kernel 2: fused per-(b,t) row kernel.
//   - (optional) p_gen recompute if workspace unavailable
//   - head-mean + softmax over S, scale by (1-p_gen)
//   - LDS scatter-add into 32000-entry copy-prob accumulator (ds_add_f32)
//   - softmax over V fused with log(combined) output
// LDS: 32000*4 + 512*4 + 256*4 = 131,072 B  -> 2 workgroups per 320KB WGP.
// Passes 2-3 over the V row hit L2 (full final_output = 131MB < 192MB L2).
// Output stores are non-temporal so the 131MB write stream does not evict
// the softmax rows from L2 between passes.
// ---------------------------------------------------------------------------
__global__ __launch_bounds__(256, 2) void pg_main_kernel(
    const float* __restrict__ dec, const float* __restrict__ fin,
    const float* __restrict__ aw, const float* __restrict__ Wp,
    const float* __restrict__ bp, const int* __restrict__ enc,
    const float* __restrict__ pg_ws, int use_ws, float* __restrict__ out) {
  __shared__ float copybuf[V_];   // 128000 B
  __shared__ float attnbuf[S_];   //   2048 B
  __shared__ float red[256];      //   1024 B

  const int tid = threadIdx.x;
  const int bt  = blockIdx.x;       // 0..1023
  const int b   = bt / T_;
  const int t   = bt - b * T_;

  // ---- zero the copy-prob accumulator -------------------------------------
  #pragma unroll
  for (int i = 0; i < V_ / 256; ++i) copybuf[tid + i * 256] = 0.0f;

  // ---- p_gen --------------------------------------------------------------
  float p;
  if (use_ws) {
    p = pg_ws[bt];
    __syncthreads();  // copybuf zeroing visible before scatter
  } else {
    float part = 0.0f;
    const float* drow = dec + (size_t)bt * D_;
    for (int k = tid; k < D_; k += 256) part += drow[k] * Wp[k];
    float dot = blockSum256(part, red, tid);  // includes barriers
    p = 1.0f / (1.0f + __expf(-(dot + bp[0])));
  }
  const float one_minus_p = 1.0f - p;

  // ---- attention: mean over heads, softmax over S, scale ------------------
  float lmax = -3.402823466e+38f;
  float aval[S_ / 256];
  #pragma unroll
  for (int i = 0; i < S_ / 256; ++i) {
    const int s = tid + i * 256;
    const float* base = aw + ((size_t)b * H_) * (T_ * S_) + (size_t)t * S_ + s;
    float a = 0.0f;
    #pragma unroll
    for (int h = 0; h < H_; ++h) a += base[(size_t)h * (T_ * S_)];
    a *= (1.0f / (float)H_);
    aval[i] = a;
    attnbuf[s] = a;
    lmax = fmaxf(lmax, a);
  }
  const float aM = blockMax256(lmax, red, tid);
  float lsum = 0.0f;
  #pragma unroll
  for (int i = 0; i < S_ / 256; ++i) lsum += __expf(aval[i] - aM);
  const float aZ = blockSum256(lsum, red, tid);
  const float ainv = one_minus_p / aZ;
  #pragma unroll
  for (int i = 0; i < S_ / 256; ++i) {
    const int s = tid + i * 256;
    attnbuf[s] = __expf(aval[i] - aM) * ainv;  // updates[b,t,s]
  }
  __syncthreads();

  // ---- scatter-add into LDS copy row (tokens may repeat -> atomic) --------
  const int* tok = enc + (size_t)b * S_;
  #pragma unroll
  for (int i = 0; i < S_ / 256; ++i) {
    const int s = tid + i * 256;
    atomicAdd(&copybuf[tok[s]], attnbuf[s]);
  }
  __syncthreads();

  // ---- softmax over V fused with output write -----------------------------
  const size_t row = (size_t)bt * V_;
  const v4f* f4 = (const v4f*)(fin + row);
  v4f* o4 = (v4f*)(out + row);
  const int NV4 = V_ / 4;          // 8000 float4 per row
  const int IT4 = NV4 / 256;       // 31 full float4 iterations
  const int remv = IT4 * 256 * 4;  // 31744; remainder = 256 scalars

  // pass 1: row max (streams from HBM, fills L2 for passes 2-3)
  float vmax = -3.402823466e+38f;
  for (int i = 0; i < IT4; ++i) {
    v4f x = f4[tid + i * 256];
    vmax = fmaxf(vmax, fmaxf(fmaxf(x.x, x.y), fmaxf(x.z, x.w)));
  }
  vmax = fmaxf(vmax, fin[row + remv + tid]);
  const float M = blockMax256(vmax, red, tid);

  // pass 2: sum of exp (hits L2)
  float vsum = 0.0f;
  for (int i = 0; i < IT4; ++i) {
    v4f x = f4[tid + i * 256];
    vsum += __expf(x.x - M) + __expf(x.y - M) + __expf(x.z - M) + __expf(x.w - M);
  }
  vsum += __expf(fin[row + remv + tid] - M);
  const float Z = blockSum256(vsum, red, tid);
  const float pinv = p / Z;

  // pass 3: out = log(p*softmax + copy); reads from L2, NT stores to HBM
  for (int i = 0; i < IT4; ++i) {
    const int idx = tid + i * 256;
    v4f x = f4[idx];
    const int v0 = idx * 4;
    v4f r;
    r.x = __logf(__expf(x.x - M) * pinv + copybuf[v0 + 0]);
    r.y = __logf(__expf(x.y - M) * pinv + copybuf[v0 + 1]);
    r.z = __logf(__expf(x.z - M) * pinv + copybuf[v0 + 2]);
    r.w = __logf(__expf(x.w - M) * pinv + copybuf[v0 + 3]);
    __builtin_nontemporal_store(r, &o4[idx]);
  }
  {
    const int v = remv + tid;
    float r = __logf(__expf(fin[row + v] - M) * pinv + copybuf[v]);
    __builtin_nontemporal_store(r, &out[row + v]);
  }
}

// ---------------------------------------------------------------------------
extern "C" void kernel_launch(void* const* d_in, const int* in_sizes, int n_in,
                              void* d_out, int out_size, void* d_ws, size_t ws_size,
                              hipStream_t stream) {
  const float* dec = (const float*)d_in[0];  // (B,T,D)
  const float* fin = (const float*)d_in[1];  // (B,T,V)
  const float* aw  = (const float*)d_in[2];  // (B,H,T,S)
  const float* Wp  = (const float*)d_in[3];  // (D,1)
  const float* bp  = (const float*)d_in[4];  // (1,)
  const int*   enc = (const int*)d_in[5];    // (B,S)
  float* out = (float*)d_out;                // (B,T,V) float32

  float* pg = (float*)d_ws;
  const int use_ws = (ws_size >= (size_t)BT_ * sizeof(float)) ? 1 : 0;

  if (use_ws) {
    pgen_wmma_kernel<<<BT_ / 16, 32, 0, stream>>>(dec, Wp, bp, pg);
  }
  pg_main_kernel<<<BT_, 256, 0, stream>>>(dec, fin, aw, Wp, bp, enc, pg, use_ws, out);
}